// _MTCNN_38250978738628
// MI455X (gfx1250) — compile-verified
//
#include <hip/hip_runtime.h>
#include <hip/hip_bf16.h>
#include <stdint.h>

#define NTOT 8192
#define NEGV (-1e30f)

typedef __attribute__((ext_vector_type(2))) float v2f;
typedef __attribute__((ext_vector_type(8))) float v8f;

__device__ __forceinline__ void boxcoords(int n, float& x1, float& y1,
                                          float& x2, float& y2, int& b) {
  b = n >> 10;                 // image index (H*W = 1024)
  int rem = n & 1023;
  int yi = rem >> 5;
  int xi = rem & 31;
  float bx = (float)xi, by = (float)yi;
  x1 = floorf((2.0f * bx + 1.0f)  / 0.3f);
  y1 = floorf((2.0f * by + 1.0f)  / 0.3f);
  x2 = floorf((2.0f * bx + 12.0f) / 0.3f);
  y2 = floorf((2.0f * by + 12.0f) / 0.3f);
}

__global__ void k_init(unsigned* mc, int* keep) {
  int i = blockIdx.x * blockDim.x + threadIdx.x;
  if (i < NTOT) keep[i] = 0;
  if (i == 0) *mc = 0u;
}

// max over valid boxes of all 4 coords (coords are non-negative -> uint-bit max works)
__global__ void k_maxcoord(const float* probs, unsigned* mc) {
  int i = blockIdx.x * blockDim.x + threadIdx.x;
  if (i >= NTOT) return;
  float x1, y1, x2, y2; int b;
  boxcoords(i, x1, y1, x2, y2, b);
  float s = probs[i];
  float m = (s >= 0.6f) ? fmaxf(fmaxf(x1, y1), fmaxf(x2, y2)) : 0.0f;
  atomicMax(mc, __float_as_uint(m));
}

// offset boxes per image, init live scores
__global__ void k_prep(const float* probs, const unsigned* mc,
                       float* gx1, float* gy1, float* gx2, float* gy2,
                       float* glive) {
  int i = blockIdx.x * blockDim.x + threadIdx.x;
  if (i >= NTOT) return;
  float x1, y1, x2, y2; int b;
  boxcoords(i, x1, y1, x2, y2, b);
  float mcv = __uint_as_float(*mc);
  float off = (float)b * (mcv + 1.0f);
  gx1[i] = x1 + off;
  gy1[i] = y1 + off;
  gx2[i] = x2 + off;
  gy2[i] = y2 + off;
  float s = probs[i];
  glive[i] = (s >= 0.6f) ? s : NEGV;
}

// Sequential greedy NMS: one workgroup, live + 3 coord arrays staged in LDS via
// CDNA5 async global->LDS loads (y2 read from L2 to stay under LDS budget).
__global__ __launch_bounds__(1024)
void k_nms(const float* gx1, const float* gy1, const float* gx2, const float* gy2,
           const float* glive, int* keep) {
  __shared__ float s_x1[NTOT];
  __shared__ float s_y1[NTOT];
  __shared__ float s_x2[NTOT];
  __shared__ float s_live[NTOT];
  __shared__ float rv[32];
  __shared__ int   ri[32];
  __shared__ float s_ax1, s_ay1, s_ax2, s_ay2, s_aarea;
  __shared__ int   s_ok;

  const int tid = threadIdx.x;

  // ---- CDNA5 async tensor-path staging: global -> LDS, tracked by ASYNCcnt ----
  for (int j = tid; j < NTOT; j += 1024) {
    asm volatile("global_load_async_to_lds_b32 %0, %1, off"
                 :: "v"((unsigned)(uintptr_t)&s_x1[j]),  "v"(gx1 + j)   : "memory");
    asm volatile("global_load_async_to_lds_b32 %0, %1, off"
                 :: "v"((unsigned)(uintptr_t)&s_y1[j]),  "v"(gy1 + j)   : "memory");
    asm volatile("global_load_async_to_lds_b32 %0, %1, off"
                 :: "v"((unsigned)(uintptr_t)&s_x2[j]),  "v"(gx2 + j)   : "memory");
    asm volatile("global_load_async_to_lds_b32 %0, %1, off"
                 :: "v"((unsigned)(uintptr_t)&s_live[j]), "v"(glive + j) : "memory");
  }
  asm volatile("s_wait_asynccnt 0" ::: "memory");
  __syncthreads();

  for (int iter = 0; iter < NTOT; ++iter) {
    // block-wide argmax of live (first-index tie-break)
    float bv = NEGV;
    int bi_ = 0;
#pragma unroll
    for (int k = 0; k < 8; ++k) {
      int j = tid + k * 1024;
      float v = s_live[j];
      if (v > bv) { bv = v; bi_ = j; }
    }
#pragma unroll
    for (int off = 16; off > 0; off >>= 1) {
      float ov = __shfl_down(bv, off);
      int   oi = __shfl_down(bi_, off);
      if (ov > bv || (ov == bv && oi < bi_)) { bv = ov; bi_ = oi; }
    }
    if ((tid & 31) == 0) { rv[tid >> 5] = bv; ri[tid >> 5] = bi_; }
    __syncthreads();
    if (tid < 32) {
      bv = rv[tid]; bi_ = ri[tid];
#pragma unroll
      for (int off = 16; off > 0; off >>= 1) {
        float ov = __shfl_down(bv, off);
        int   oi = __shfl_down(bi_, off);
        if (ov > bv || (ov == bv && oi < bi_)) { bv = ov; bi_ = oi; }
      }
      if (tid == 0) {
        int ok = bv > (NEGV * 0.5f);
        s_ok = ok;
        if (ok) {
          keep[bi_] = 1;
          float ax1 = s_x1[bi_], ay1 = s_y1[bi_];
          float ax2 = s_x2[bi_], ay2 = gy2[bi_];
          s_ax1 = ax1; s_ay1 = ay1; s_ax2 = ax2; s_ay2 = ay2;
          s_aarea = (ax2 - ax1 + 1.0f) * (ay2 - ay1 + 1.0f);
        }
      }
    }
    __syncthreads();
    if (!s_ok) break;  // remaining reference iterations are keep-no-ops

    const float ax1 = s_ax1, ay1 = s_ay1, ax2 = s_ax2, ay2 = s_ay2, aarea = s_aarea;
#pragma unroll
    for (int k = 0; k < 8; ++k) {
      int j = tid + k * 1024;
      float jx1 = s_x1[j], jy1 = s_y1[j], jx2 = s_x2[j], jy2 = gy2[j];
      float xx1 = fmaxf(ax1, jx1), yy1 = fmaxf(ay1, jy1);
      float xx2 = fminf(ax2, jx2), yy2 = fminf(ay2, jy2);
      float inter = fmaxf(0.0f, xx2 - xx1 + 1.0f) * fmaxf(0.0f, yy2 - yy1 + 1.0f);
      float areaj = (jx2 - jx1 + 1.0f) * (jy2 - jy1 + 1.0f);
      float iou = inter / (aarea + areaj - inter);
      if (iou > 0.5f) s_live[j] = NEGV;   // includes self (IoU == 1)
    }
    __syncthreads();
  }
}

// Box refinement + output assembly. Columns 0..3 of out go through
// V_WMMA_F32_16X16X4_F32 with an identity B (exact in f32): D = A*I + 0.
// A layout (16x4 f32): lanes 0-15 supply K=0 (VGPR0) / K=1 (VGPR1);
// lanes 16-31 supply K=2 / K=3 for the same rows M = lane&15.
__global__ __launch_bounds__(32)
void k_out(const float* probs, const float* reg, const int* keep, float* out) {
  const int lane = threadIdx.x;
  const int base = blockIdx.x * 16;
  const int m = lane & 15;
  const int n = base + m;

  float x1, y1, x2, y2; int b;
  boxcoords(n, x1, y1, x2, y2, b);
  int rem = n & 1023;
  int yi = rem >> 5, xi = rem & 31;
  const int rb = b * 4096 + yi * 32 + xi;   // reg[b][c][yi][xi], c-stride 1024
  float r0 = reg[rb], r1 = reg[rb + 1024], r2 = reg[rb + 2048], r3 = reg[rb + 3072];

  float w = x2 - x1, h = y2 - y1;
  float qq1 = x1 + r0 * w, qq2 = y1 + r1 * h;
  float qq3 = x2 + r2 * w, qq4 = y2 + r3 * h;
  float hh = qq4 - qq2, ww = qq3 - qq1;
  float l = fmaxf(ww, hh);
  float rx1 = qq1 + ww * 0.5f - l * 0.5f;
  float ry1 = qq2 + hh * 0.5f - l * 0.5f;
  float rx2 = rx1 + l, ry2 = ry1 + l;

  float s  = probs[n];
  float kf = keep[n] ? 1.0f : 0.0f;

  // A tile: K=0 -> rx1, K=1 -> ry1 (lanes 0-15); K=2 -> rx2, K=3 -> ry2 (lanes 16-31)
  float a0 = (lane < 16) ? rx1 * kf : rx2 * kf;
  float a1 = (lane < 16) ? ry1 * kf : ry2 * kf;
  // B identity (4x16): B[k][n] = (k==n). VGPR0 holds K=0 (lanes 0-15) / K=2 (16-31),
  // VGPR1 holds K=1 / K=3.
  float b0 = (lane == 0 || lane == 18) ? 1.0f : 0.0f;
  float b1 = (lane == 1 || lane == 19) ? 1.0f : 0.0f;

  v2f A  = {a0, a1};
  v2f Bm = {b0, b1};
  v8f C  = {};
  v8f D = __builtin_amdgcn_wmma_f32_16x16x4_f32(false, A, false, Bm,
                                                (short)0, C, false, false);

  // D layout: VGPR r = row M=r (lanes 0-15, N=lane) / M=8+r (lanes 16-31, N=lane-16)
  int col = (lane < 4) ? lane : ((lane >= 16 && lane < 20) ? (lane - 16) : -1);
  int rowbase = (lane < 16) ? 0 : 8;
  if (col >= 0) {
#pragma unroll
    for (int r = 0; r < 8; ++r)
      out[(base + rowbase + r) * 5 + col] = D[r];
  }
  if (lane < 16) out[n * 5 + 4] = s * kf;   // score column
}

extern "C" void kernel_launch(void* const* d_in, const int* in_sizes, int n_in,
                              void* d_out, int out_size, void* d_ws, size_t ws_size,
                              hipStream_t stream) {
  const float* probs = (const float*)d_in[0];
  const float* reg   = (const float*)d_in[1];
  float* out = (float*)d_out;

  float* Wf = (float*)d_ws;
  unsigned* mc = (unsigned*)d_ws;       // [0] : max_coord bits
  float* gx1   = Wf + 64;
  float* gy1   = gx1 + NTOT;
  float* gx2   = gy1 + NTOT;
  float* gy2   = gx2 + NTOT;
  float* glive = gy2 + NTOT;
  int*   gkeep = (int*)(glive + NTOT);

  k_init    <<<(NTOT + 255) / 256, 256, 0, stream>>>(mc, gkeep);
  k_maxcoord<<<(NTOT + 255) / 256, 256, 0, stream>>>(probs, mc);
  k_prep    <<<(NTOT + 255) / 256, 256, 0, stream>>>(probs, mc, gx1, gy1, gx2, gy2, glive);
  k_nms     <<<1, 1024, 0, stream>>>(gx1, gy1, gx2, gy2, glive, gkeep);
  k_out     <<<NTOT / 16, 32, 0, stream>>>(probs, reg, gkeep, out);
}